// SSIMLoss_25374666785375
// MI455X (gfx1250) — compile-verified
//
#include <hip/hip_runtime.h>
#include <math.h>

// ---------------------------------------------------------------------------
// SSIM loss, fused, CDNA5 (gfx1250).
//  - 5 depthwise 11x11 Gaussian convs as two banded-matrix passes through
//    V_WMMA_F32_16X16X4_F32 (full fp32: sigma = E[x^2]-mu^2 cancellation
//    rules out bf16/f16 inputs).
//  - Double-buffered GLOBAL_LOAD_ASYNC_TO_LDS prefetch of the next halo tile
//    while WMMAs process the current one (ASYNCcnt / s_wait_asynccnt).
//  - Deterministic two-stage mean reduction (no atomics).
// ---------------------------------------------------------------------------

typedef float v8f __attribute__((ext_vector_type(8)));
typedef float v2f __attribute__((ext_vector_type(2)));

#define IMG_H 512
#define IMG_W 512
#define N_IMG 32
#define TILE 16
#define TOTAL_TILES (N_IMG * (IMG_H / TILE) * (IMG_W / TILE))  // 32768
#define TILES_PER_WAVE 16
#define N_BLOCKS (TOTAL_TILES / TILES_PER_WAVE)                // 2048 partials
#define C1 (0.01f * 0.01f)
#define C2 (0.03f * 0.03f)
#define KIN 28   // 26 halo rows padded to 7 K-steps of 4

#define AS1 __attribute__((address_space(1)))
#define AS3 __attribute__((address_space(3)))

// ---- async global -> LDS copy of one dword (per-lane, EXEC-masked) --------
__device__ __forceinline__ void async_copy_b32(const float* gsrc, float* ldst) {
#if __has_builtin(__builtin_amdgcn_global_load_async_to_lds_b32)
    __builtin_amdgcn_global_load_async_to_lds_b32(
        (AS1 int*)(unsigned long long)(uintptr_t)gsrc,
        (AS3 int*)(unsigned)(uintptr_t)ldst, 0, 0);
#else
    // GV mode: vdst = LDS byte address (addr[31:0] of the flat pointer),
    // vaddr = 64-bit global address, SADDR = off.
    unsigned lds_addr = (unsigned)(uintptr_t)ldst;
    unsigned long long gaddr = (unsigned long long)(uintptr_t)gsrc;
    asm volatile("global_load_async_to_lds_b32 %0, %1, off"
                 :: "v"(lds_addr), "v"(gaddr) : "memory");
#endif
}

__device__ __forceinline__ void wait_async0() {
#if __has_builtin(__builtin_amdgcn_s_wait_asynccnt)
    __builtin_amdgcn_s_wait_asynccnt(0);
#else
    asm volatile("s_wait_asynccnt 0x0" ::: "memory");
#endif
}

__global__ __launch_bounds__(32)
void ssim_tiles_kernel(const float* __restrict__ img1,
                       const float* __restrict__ img2,
                       float* __restrict__ partials)
{
    // Per-wave (blockDim == 32) LDS: double-buffered inputs + vertical stage.
    __shared__ float sIn1[2][KIN][32];
    __shared__ float sIn2[2][KIN][32];
    __shared__ float sVert[5][TILE][32];
    __shared__ float sG[16];

    const int lane = threadIdx.x;            // 0..31, wave32
    const int m    = lane & 15;              // M (out row) / N (out col) index
    const int koff = (lane < 16) ? 0 : 2;    // K sub-offset per half-wave

    // Gaussian taps in double (matches numpy) then cast to f32.
    {
        double s = 0.0;
        #pragma unroll
        for (int i = 0; i < 11; ++i) {
            double d = (double)(i - 5);
            s += exp(-(d * d) / 4.5);
        }
        float gv = 0.0f;
        if (lane < 11) {
            double d = (double)(lane - 5);
            gv = (float)(exp(-(d * d) / 4.5) / s);
        }
        if (lane < 16) sG[lane] = gv;
    }
    __syncthreads();

    auto wv = [&](int d) -> float {
        return (d >= 0 && d <= 10) ? sG[d] : 0.0f;
    };

    // Banded weight fragments (constant across tiles).
    v2f aW[7], bW[7];
    #pragma unroll
    for (int kk = 0; kk < 7; ++kk) {
        const int r0 = kk * 4 + koff;
        aW[kk].x = wv(r0 + 0 - m);   // vertical A: Wv[M][K] = g[K - M]
        aW[kk].y = wv(r0 + 1 - m);
        bW[kk].x = wv(r0 + 0 - m);   // horizontal B: Wh[K][N] = g[K - N]
        bW[kk].y = wv(r0 + 1 - m);
    }

    // Issue the async fill for tile (blk, t) into buffer buf.
    auto issue_tile_loads = [&](int t, int buf) {
        const int tile = blockIdx.x * TILES_PER_WAVE + t;
        const int img  = tile >> 10;
        const int rem  = tile & 1023;
        const int y0   = ((rem >> 5) << 4) - 5;   // halo origin
        const int x0   = ((rem & 31) << 4) - 5;
        const float* __restrict__ p1 = img1 + (size_t)img * IMG_H * IMG_W;
        const float* __restrict__ p2 = img2 + (size_t)img * IMG_H * IMG_W;
        #pragma unroll 4
        for (int r = 0; r < KIN; ++r) {
            const int row = y0 + r;
            const int col = x0 + lane;
            const bool ok = (r < 26) && (lane < 26) &&
                            (row >= 0) && (row < IMG_H) &&
                            (col >= 0) && (col < IMG_W);
            const int idx = row * IMG_W + col;
            if (ok) {
                async_copy_b32(p1 + idx, &sIn1[buf][r][lane]);
                async_copy_b32(p2 + idx, &sIn2[buf][r][lane]);
            } else {
                sIn1[buf][r][lane] = 0.0f;   // zero padding (ds_store path)
                sIn2[buf][r][lane] = 0.0f;
            }
        }
    };

    float lsum = 0.0f;

    issue_tile_loads(0, 0);                  // prologue prefetch

    #pragma unroll 1
    for (int t = 0; t < TILES_PER_WAVE; ++t) {
        const int cur = t & 1;

        wait_async0();                       // current buffer resident
        __syncthreads();                     // + dscnt fence for zero-pads

        if (t + 1 < TILES_PER_WAVE)
            issue_tile_loads(t + 1, cur ^ 1);   // overlap with WMMA below

        const float (* __restrict__ in1)[32] = sIn1[cur];
        const float (* __restrict__ in2)[32] = sIn2[cur];

        // ---- vertical blur of 5 fields: D = Wv[16x28] * Field[28x16] ----
        #pragma unroll
        for (int ct = 0; ct < 2; ++ct) {
            v8f acc[5] = {};
            const int c = ct * 16 + m;
            #pragma unroll
            for (int kk = 0; kk < 7; ++kk) {
                const int r0 = kk * 4 + koff;
                const float i1a = in1[r0 + 0][c];
                const float i1b = in1[r0 + 1][c];
                const float i2a = in2[r0 + 0][c];
                const float i2b = in2[r0 + 1][c];
                v2f b0 = {i1a,       i1b      };
                v2f b1 = {i2a,       i2b      };
                v2f b2 = {i1a * i1a, i1b * i1b};
                v2f b3 = {i2a * i2a, i2b * i2b};
                v2f b4 = {i1a * i2a, i1b * i2b};
                acc[0] = __builtin_amdgcn_wmma_f32_16x16x4_f32(false, aW[kk], false, b0, (short)0, acc[0], false, false);
                acc[1] = __builtin_amdgcn_wmma_f32_16x16x4_f32(false, aW[kk], false, b1, (short)0, acc[1], false, false);
                acc[2] = __builtin_amdgcn_wmma_f32_16x16x4_f32(false, aW[kk], false, b2, (short)0, acc[2], false, false);
                acc[3] = __builtin_amdgcn_wmma_f32_16x16x4_f32(false, aW[kk], false, b3, (short)0, acc[3], false, false);
                acc[4] = __builtin_amdgcn_wmma_f32_16x16x4_f32(false, aW[kk], false, b4, (short)0, acc[4], false, false);
            }
            const int rbase = (lane < 16) ? 0 : 8;
            #pragma unroll
            for (int f = 0; f < 5; ++f) {
                #pragma unroll
                for (int v = 0; v < 8; ++v) {
                    sVert[f][v + rbase][ct * 16 + m] = acc[f][v];
                }
            }
        }
        __syncthreads();

        // ---- horizontal blur: D = Vert[16x28] * Wh[28x16] ----
        v8f blur[5] = {};
        #pragma unroll
        for (int f = 0; f < 5; ++f) {
            #pragma unroll
            for (int kk = 0; kk < 7; ++kk) {
                const int c0 = kk * 4 + koff;
                v2f a;
                a.x = sVert[f][m][c0 + 0];
                a.y = sVert[f][m][c0 + 1];
                blur[f] = __builtin_amdgcn_wmma_f32_16x16x4_f32(false, a, false, bW[kk], (short)0, blur[f], false, false);
            }
        }

        // ---- SSIM map on the 8 accumulator elements ----
        #pragma unroll
        for (int v = 0; v < 8; ++v) {
            const float mu1 = blur[0][v];
            const float mu2 = blur[1][v];
            const float m1s = mu1 * mu1;
            const float m2s = mu2 * mu2;
            const float m12 = mu1 * mu2;
            const float s11 = blur[2][v] - m1s;
            const float s22 = blur[3][v] - m2s;
            const float s12 = blur[4][v] - m12;
            const float num = (2.0f * m12 + C1) * (2.0f * s12 + C2);
            const float den = (m1s + m2s + C1) * (s11 + s22 + C2);
            lsum += num / den;
        }

        __syncthreads();   // sVert reuse next iteration
    }

    // Wave32 shuffle reduction (fixed order -> deterministic).
    #pragma unroll
    for (int sh = 16; sh > 0; sh >>= 1) {
        lsum += __shfl_xor(lsum, sh, 32);
    }
    if (lane == 0) partials[blockIdx.x] = lsum;
}

__global__ __launch_bounds__(256)
void ssim_reduce_kernel(const float* __restrict__ partials,
                        float* __restrict__ out)
{
    __shared__ double sd[256];
    const int tid = threadIdx.x;
    double a = 0.0;
    for (int i = tid; i < N_BLOCKS; i += 256) a += (double)partials[i];
    sd[tid] = a;
    __syncthreads();
    for (int s = 128; s > 0; s >>= 1) {
        if (tid < s) sd[tid] += sd[tid + s];
        __syncthreads();
    }
    if (tid == 0) {
        const double npix = (double)N_IMG * IMG_H * IMG_W;  // 8388608
        out[0] = (float)(1.0 - sd[0] / npix);
    }
}

extern "C" void kernel_launch(void* const* d_in, const int* in_sizes, int n_in,
                              void* d_out, int out_size, void* d_ws, size_t ws_size,
                              hipStream_t stream) {
    const float* img1 = (const float*)d_in[0];
    const float* img2 = (const float*)d_in[1];
    float* out      = (float*)d_out;
    float* partials = (float*)d_ws;   // N_BLOCKS floats = 8 KB

    ssim_tiles_kernel<<<N_BLOCKS, 32, 0, stream>>>(img1, img2, partials);
    ssim_reduce_kernel<<<1, 256, 0, stream>>>(partials, out);
}